// CrossmotionModule_80418967650448
// MI455X (gfx1250) — compile-verified
//
#include <hip/hip_runtime.h>
#include <hip/hip_bf16.h>

// Crossmotion collapse: the reference is a purely linear map of a rank-3
// (per b,t) pairwise tensor, so the O(M^2) GEMMs collapse to O(M) vectors.
//   out[bt,m,o] = vis'_m*(cx_m*P0[o] + cy_m*P1[o] - PS[o]) + Q[m,o]
// P* come from fp32 WMMA GEMMs (V_WMMA_F32_16X16X4_F32); the final pass is
// HBM-write-bound (~201 MB -> ~9us @ 23.3 TB/s) and uses b128 NT stores.

#define B_  4
#define M_  256
#define T_  64
#define DM  512   // D_MOT (= 2*M = fc1 in/out)
#define DO  768   // D_OUT

typedef float v2f __attribute__((ext_vector_type(2)));
typedef float v4f __attribute__((ext_vector_type(4)));
typedef float v8f __attribute__((ext_vector_type(8)));

__device__ __forceinline__ float fixnum(float x) {
    // jnp.nan_to_num: nan->0, +inf->FLT_MAX, -inf->-FLT_MAX
    if (x != x) return 0.0f;
    if (x ==  __builtin_inff()) return  3.402823466e38f;
    if (x == -__builtin_inff()) return -3.402823466e38f;
    return x;
}

__device__ __forceinline__ v8f wmma_f32(v2f a, v2f b, v8f c) {
    // (neg_a, A, neg_b, B, c_mod, C, reuse_a, reuse_b)
    return __builtin_amdgcn_wmma_f32_16x16x4_f32(false, a, false, b, (short)0, c,
                                                 false, false);
}

// ---- K0: Qbase[o] = b2[o] + sum_d b1[d]*W2[d,o]  (768 outputs) ----
__global__ void k0_qbase(const float* __restrict__ b1, const float* __restrict__ w2,
                         const float* __restrict__ b2, float* __restrict__ qbase) {
    int o = blockIdx.x * blockDim.x + threadIdx.x;
    float s = b2[o];
    for (int d = 0; d < DM; ++d) s += b1[d] * w2[d * DO + o];
    qbase[o] = s;
}

// ---- K1: Q[m,o] = sum_d pos[m,d]*W2[512+d,o] + Qbase[o]   (256x768, K=512) ----
// one wave (32 threads) per 16x16 tile, fp32 WMMA, K-step 4
__global__ void k1_q(const float* __restrict__ pos, const float* __restrict__ w2,
                     const float* __restrict__ qbase, float* __restrict__ q) {
    const int lane = threadIdx.x, l15 = lane & 15, half = lane >> 4;
    const int rowA = blockIdx.y * 16 + l15;   // m
    const int colB = blockIdx.x * 16 + l15;   // o
    v8f acc = {};
    for (int k = 0; k < DM; k += 4) {
        const int k0 = k + half * 2;
        v2f a, b;
        a.x = pos[rowA * DM + k0];
        a.y = pos[rowA * DM + k0 + 1];
        b.x = w2[(DM + k0) * DO + colB];
        b.y = w2[(DM + k0 + 1) * DO + colB];
        acc = wmma_f32(a, b, acc);
    }
    const float qb = qbase[colB];
    #pragma unroll
    for (int v = 0; v < 8; ++v) {
        const int row = blockIdx.y * 16 + v + half * 8;
        q[row * DO + colB] = acc[v] + qb;
    }
}

// ---- K2: per bt, AS rows {A0,A1,S} over d  -> AS is a (768 x 512) row-major
//          matrix with row r = bt*3 + j. NaN/vis fixups folded into A-frags. ----
__global__ void k2_as(const float* __restrict__ coords, const float* __restrict__ vis,
                      const float* __restrict__ w1, float* __restrict__ as_) {
    const int lane = threadIdx.x, l15 = lane & 15, half = lane >> 4;
    const int btRow = blockIdx.y * 16 + l15;          // A-frag row = bt
    const int b = btRow >> 6, t = btRow & 63;
    const int dcol = blockIdx.x * 16 + l15;           // B-frag col = d
    const float* cb = coords + b * (M_ * T_ * 2) + t * 2;  // + n*T*2 (+k)
    const float* vb = vis    + b * (M_ * T_)     + t;      // + n*T
    v8f acc0 = {}, acc1 = {}, accS = {};
    for (int n = 0; n < M_; n += 4) {
        const int k0 = n + half * 2;
        v2f av, ax, ay;
        #pragma unroll
        for (int j = 0; j < 2; ++j) {
            const int kk = k0 + j;
            const float cxr = cb[kk * (T_ * 2) + 0];
            const float cyr = cb[kk * (T_ * 2) + 1];
            const float vv  = vb[kk * T_];
            const float vf  = (cxr != cxr) ? 0.0f : vv;   // vis'=0 where x is NaN
            av[j] = vf;
            ax[j] = vf * fixnum(cxr);
            ay[j] = vf * fixnum(cyr);
        }
        v2f b0, b1;   // W1 even rows / odd rows, consecutive K in .x/.y
        b0.x = w1[(2 * k0)     * DM + dcol];
        b0.y = w1[(2 * k0 + 2) * DM + dcol];
        b1.x = w1[(2 * k0 + 1) * DM + dcol];
        b1.y = w1[(2 * k0 + 3) * DM + dcol];
        acc0 = wmma_f32(av, b0, acc0);   // A0 += vis (x) W1_even
        acc1 = wmma_f32(av, b1, acc1);   // A1 += vis (x) W1_odd
        accS = wmma_f32(ax, b0, accS);   // S  += vis*cx (x) W1_even
        accS = wmma_f32(ay, b1, accS);   //    += vis*cy (x) W1_odd
    }
    #pragma unroll
    for (int v = 0; v < 8; ++v) {
        const int bt = blockIdx.y * 16 + v + half * 8;
        float* p = as_ + (bt * 3) * DM + dcol;   // r = bt*3 + j
        p[0]      = acc0[v];
        p[DM]     = acc1[v];
        p[2 * DM] = accS[v];
    }
}

// ---- K3: P = AS(768x512) @ W2[:512](512x768)  -> P row r = bt*3+j ----
__global__ void k3_p(const float* __restrict__ as_, const float* __restrict__ w2,
                     float* __restrict__ p) {
    const int lane = threadIdx.x, l15 = lane & 15, half = lane >> 4;
    const int rowA = blockIdx.y * 16 + l15;
    const int colB = blockIdx.x * 16 + l15;
    v8f acc = {};
    for (int k = 0; k < DM; k += 4) {
        const int k0 = k + half * 2;
        v2f a, b;
        a.x = as_[rowA * DM + k0];
        a.y = as_[rowA * DM + k0 + 1];
        b.x = w2[k0 * DO + colB];
        b.y = w2[(k0 + 1) * DO + colB];
        acc = wmma_f32(a, b, acc);
    }
    #pragma unroll
    for (int v = 0; v < 8; ++v) {
        const int row = blockIdx.y * 16 + v + half * 8;
        p[row * DO + colB] = acc[v];
    }
}

// ---- K4: bandwidth-bound broadcast. 192 threads = one float4 column slice
//      each; P0/P1/PS are hoisted to registers (LDS only as a staging hop);
//      inner loop = broadcast scalar loads + b128 Q load + b128 NT store.
//      out[bt,m,o] = vis'*(cx*P0[o] + cy*P1[o] - PS[o]) + Q[m,o]
__global__ __launch_bounds__(192) void k4_out(
        const float* __restrict__ coords, const float* __restrict__ vis,
        const float* __restrict__ p, const float* __restrict__ q,
        float* __restrict__ out) {
    __shared__ float sP[3 * DO];                 // 9 KB of 320 KB WGP LDS
    const int bt = blockIdx.y, chunk = blockIdx.x, tid = threadIdx.x;
    {   // stage 2304 floats = 576 float4 with 192 threads (3 iters)
        const v4f* src = (const v4f*)(p + bt * (3 * DO));
        v4f* dst = (v4f*)sP;
        #pragma unroll
        for (int i = 0; i < 3; ++i) dst[tid + i * 192] = src[tid + i * 192];
    }
    __syncthreads();
    const int o = tid * 4;
    const v4f p0 = *(const v4f*)(sP + o);            // loop-invariant
    const v4f p1 = *(const v4f*)(sP + DO + o);
    const v4f ps = *(const v4f*)(sP + 2 * DO + o);
    const int b = bt >> 6, t = bt & 63;
    const float* cb = coords + b * (M_ * T_ * 2) + t * 2;
    const float* vb = vis    + b * (M_ * T_)     + t;
    for (int mi = 0; mi < 32; ++mi) {
        const int m = chunk * 32 + mi;
        const float cxr = cb[m * (T_ * 2)];          // broadcast (same addr/wave)
        const float cyr = cb[m * (T_ * 2) + 1];
        const float vv  = vb[m * T_];
        const float vf  = (cxr != cxr) ? 0.0f : vv;
        const float cx = fixnum(cxr), cy = fixnum(cyr);
        const v4f qv = *(const v4f*)(q + m * DO + o);
        if (mi + 1 < 32) __builtin_prefetch(q + (m + 1) * DO + o, 0, 1);
        v4f r;
        r.x = vf * (cx * p0.x + cy * p1.x - ps.x) + qv.x;
        r.y = vf * (cx * p0.y + cy * p1.y - ps.y) + qv.y;
        r.z = vf * (cx * p0.z + cy * p1.z - ps.z) + qv.z;
        r.w = vf * (cx * p0.w + cy * p1.w - ps.w) + qv.w;
        __builtin_nontemporal_store(r, (v4f*)(out + (size_t)(bt * M_ + m) * DO + o));
    }
}

extern "C" void kernel_launch(void* const* d_in, const int* in_sizes, int n_in,
                              void* d_out, int out_size, void* d_ws, size_t ws_size,
                              hipStream_t stream) {
    const float* coords = (const float*)d_in[0];  // (B,M,T,2)
    const float* vis    = (const float*)d_in[1];  // (B,M,T)
    const float* pos    = (const float*)d_in[2];  // (M,512)
    const float* w1     = (const float*)d_in[3];  // (512,512)
    const float* b1     = (const float*)d_in[4];  // (512,)
    const float* w2     = (const float*)d_in[5];  // (1024,768)
    const float* b2     = (const float*)d_in[6];  // (768,)
    float* out = (float*)d_out;

    float* ws    = (float*)d_ws;
    float* qbase = ws;                    //    768 floats
    float* q     = ws + 1024;             // 196608 floats (256x768)
    float* as_   = q + 196608;            // 393216 floats (768x512)
    float* p     = as_ + 393216;          // 589824 floats (768x768)
    // total ~4.6 MB of workspace

    k0_qbase<<<dim3(3),      dim3(256), 0, stream>>>(b1, w2, b2, qbase);
    k1_q    <<<dim3(48, 16), dim3(32),  0, stream>>>(pos, w2, qbase, q);
    k2_as   <<<dim3(32, 16), dim3(32),  0, stream>>>(coords, vis, w1, as_);
    k3_p    <<<dim3(48, 48), dim3(32),  0, stream>>>(as_, w2, p);
    k4_out  <<<dim3(8, 256), dim3(192), 0, stream>>>(coords, vis, p, q, out);
}